// GAT_7499012899343
// MI455X (gfx1250) — compile-verified
//
#include <hip/hip_runtime.h>

// GAT forward for MI455X (gfx1250), wave32, f32 WMMA 16x16x4.
// B=4, N=2048, Cin=256, F=64, H=4. Softmax is over the *i* axis (reference
// quirk) -> att[i,j] = exp(e[i,j]-m[j]) * rd[j] with column stats only.
// We fold rd[j] into the rows of h (G = h*rd) and compute hp = P @ G with
// WMMA, never materializing the 256MB attention matrix.
//
// Precision choice: f32 WMMA. Total math ~10 GFLOP is negligible on any WMMA
// ceiling; the bottleneck is masked-exp generation (VALU/trans) + adj L2
// traffic, which low precision would not improve. So keep exact f32.

#define ALPHA 0.2f
#define NEGV  -1e12f

typedef __attribute__((ext_vector_type(2))) float v2f;
typedef __attribute__((ext_vector_type(8))) float v8f;

static constexpr int B_  = 4;
static constexpr int N_  = 2048;
static constexpr int CIN = 256;
static constexpr int F_  = 64;
static constexpr int H_  = 4;
static constexpr int BH  = 16;
static constexpr int LDT = 68;   // LDS tile stride: 68 % 64 == 4 -> b64 loads conflict-free, 8B aligned

__device__ __forceinline__ v8f wmma_f32_16x16x4(v2f a, v2f b, v8f c) {
  // (neg_a, A, neg_b, B, c_mod, C, reuse_a, reuse_b)
  return __builtin_amdgcn_wmma_f32_16x16x4_f32(false, a, false, b, (short)0, c,
                                               false, false);
}

// ---- CDNA5 async global->LDS path via inline asm (portable across clang
// builtin signature differences; see CDNA5 ISA). ----
#if defined(__gfx1250__)
#define USE_ASYNC_G 1
__device__ __forceinline__ void async_copy16(const float* gsrc, float* lds_dst) {
  // Generic pointer to __shared__ is {aperture_hi, lds_byte_offset}; the
  // instruction's VDST wants the per-lane LDS byte offset (low 32 bits).
  const unsigned lds_off = (unsigned)(size_t)lds_dst;
  const unsigned long long ga = (unsigned long long)(size_t)gsrc;
  asm volatile("global_load_async_to_lds_b128 %0, %1, off"
               :: "v"(lds_off), "v"(ga) : "memory");
}
#if __has_builtin(__builtin_amdgcn_s_wait_asynccnt)
#define WAIT_ASYNC() __builtin_amdgcn_s_wait_asynccnt(0)
#else
#define WAIT_ASYNC() asm volatile("s_wait_asynccnt 0x0" ::: "memory")
#endif
#else
#define USE_ASYNC_G 0
#endif

// ---------------------------------------------------------------------------
// Kernel 1: h[bh][n][f] = x[b] @ W[hd]  (per-head projection GEMM)
// 256 threads = 8 waves; 64x64 block tile, K chunks of 64.
// wave w -> i-tile w&3, f-tiles {w>>2, (w>>2)+2}.
// ---------------------------------------------------------------------------
__global__ __launch_bounds__(256) void k_proj(const float* __restrict__ x,
                                              const float* __restrict__ W,
                                              float* __restrict__ h) {
  __shared__ float As[64 * LDT];
  __shared__ float Bs[64 * LDT];
  const int tid  = threadIdx.x;
  const int bh   = blockIdx.y, b = bh >> 2, hd = bh & 3;
  const int i0   = blockIdx.x * 64;
  const int wave = tid >> 5, lane = tid & 31;
  const int it   = wave & 3, ft0 = wave >> 2;
  const int mm   = lane & 15, kb = (lane >> 4) << 1;
  v8f acc0 = {}; v8f acc1 = {};

  for (int kc = 0; kc < CIN; kc += 64) {
    __syncthreads();
#pragma unroll
    for (int r = 0; r < 16; ++r) {
      const int idx = r * 256 + tid;
      const int rr = idx >> 6, cc = idx & 63;
      As[rr * LDT + cc] = x[(size_t)(b * N_ + i0 + rr) * CIN + kc + cc];
      Bs[rr * LDT + cc] = W[(size_t)(hd * CIN + kc + rr) * F_ + cc];
    }
    __syncthreads();
    const float* pa = &As[(it * 16 + mm) * LDT];
#pragma unroll
    for (int kk = 0; kk < 16; ++kk) {
      const int k0 = kk * 4 + kb;
      const v2f a = *(const v2f*)&pa[k0];                     // ds_load_b64
      v2f b0 = { Bs[k0 * LDT + ft0 * 16 + mm], Bs[(k0 + 1) * LDT + ft0 * 16 + mm] };
      v2f b1 = { Bs[k0 * LDT + (ft0 + 2) * 16 + mm],
                 Bs[(k0 + 1) * LDT + (ft0 + 2) * 16 + mm] };
      acc0 = wmma_f32_16x16x4(a, b0, acc0);
      acc1 = wmma_f32_16x16x4(a, b1, acc1);
    }
  }
  const int mbse = (lane >> 4) << 3;
#pragma unroll
  for (int r = 0; r < 8; ++r) {
    const int mrow = it * 16 + mbse + r;
    float* hrow = h + (size_t)(bh * N_ + i0 + mrow) * F_;
    hrow[ft0 * 16 + mm]       = acc0[r];
    hrow[(ft0 + 2) * 16 + mm] = acc1[r];
  }
}

// ---------------------------------------------------------------------------
// Kernel 2: f1[bh][n] = h[bh][n] . a1[hd], f2 likewise.
// ---------------------------------------------------------------------------
__global__ __launch_bounds__(256) void k_fvec(const float* __restrict__ h,
                                              const float* __restrict__ a1,
                                              const float* __restrict__ a2,
                                              float* __restrict__ f1,
                                              float* __restrict__ f2) {
  const int idx = blockIdx.x * 256 + threadIdx.x;  // bh*N + i
  const int hd  = (idx >> 11) & 3;
  const float* hp = h + (size_t)idx * F_;
  float s1 = 0.f, s2 = 0.f;
#pragma unroll
  for (int f = 0; f < F_; ++f) {
    const float v = hp[f];
    s1 += v * a1[hd * F_ + f];
    s2 += v * a2[hd * F_ + f];
  }
  f1[idx] = s1;
  f2[idx] = s2;
}

// ---------------------------------------------------------------------------
// Kernel 3: column softmax stats, SINGLE pass over adj.
// e is bounded (|f1+f2| small) so unshifted sum exp(e) cannot overflow:
//   m[j] = max_i e[i,j];  rd[j] = exp(m[j]) / sum_i exp(e[i,j])
// Masked entries give exp(-1e12) == 0. All-masked column -> sum==0 ->
// uniform 1/N with m = NEG, matching the reference softmax exactly.
// Thread handles 4 consecutive columns (int4 adj loads, fully coalesced).
// ---------------------------------------------------------------------------
__global__ __launch_bounds__(256) void k_stats(const int* __restrict__ adj,
                                               const float* __restrict__ f1,
                                               const float* __restrict__ f2,
                                               float* __restrict__ mcol,
                                               float* __restrict__ rdcol) {
  const int bh = blockIdx.y;
  const int j0 = (blockIdx.x * 256 + threadIdx.x) * 4;
  const float* f1b = f1 + bh * N_;
  const float* f2b = f2 + bh * N_;
  float f2v[4], mx[4], sm[4];
#pragma unroll
  for (int t = 0; t < 4; ++t) {
    f2v[t] = f2b[j0 + t];
    mx[t] = -3.0e38f;
    sm[t] = 0.f;
  }
  for (int i = 0; i < N_; ++i) {
    const int4 av = *(const int4*)(adj + (size_t)i * N_ + j0);
    const float f1v = f1b[i];
    const int aa[4] = {av.x, av.y, av.z, av.w};
#pragma unroll
    for (int t = 0; t < 4; ++t) {
      float v = f1v + f2v[t];
      v = fmaxf(v, ALPHA * v);                 // leaky (alpha < 1)
      const float e = aa[t] > 0 ? v : NEGV;
      mx[t] = fmaxf(mx[t], e);
      sm[t] += __expf(e);                      // exp(-1e12) == 0
    }
  }
#pragma unroll
  for (int t = 0; t < 4; ++t) {
    float m = mx[t], rdv;
    if (m <= -0.5e12f) {                       // column fully masked
      m = NEGV;
      rdv = 1.0f / (float)N_;
    } else {
      rdv = __expf(m) / sm[t];                 // 1 / sum exp(e - m)
    }
    mcol[bh * N_ + j0 + t]  = m;
    rdcol[bh * N_ + j0 + t] = rdv;
  }
}

// ---------------------------------------------------------------------------
// Kernel 3b: G[bh][j][f] = h[bh][j][f] * rd[bh][j]   (fold softmax denom)
// ---------------------------------------------------------------------------
__global__ __launch_bounds__(256) void k_gmat(const float* __restrict__ h,
                                              const float* __restrict__ rd,
                                              float* __restrict__ g) {
  const int idx = blockIdx.x * 256 + threadIdx.x;   // float4 index
  const int e4 = idx * 4;
  const int row = e4 >> 6;                          // bh*N + j
  const float4 hv = *(const float4*)&h[e4];
  const float r = rd[row];
  float4 o; o.x = hv.x * r; o.y = hv.y * r; o.z = hv.z * r; o.w = hv.w * r;
  *(float4*)&g[e4] = o;
}

// ---------------------------------------------------------------------------
// Kernel 4: fused attention GEMM. hp = P @ G:
//   P[i,j] = exp(e[i,j]-m[j])  built on the fly into LDS per 64x64 chunk,
//   G tile async-DMA'd global->LDS (overlapped with the exp VALU work).
// ReLU'd result written straight into concat layout cat[b][n][hd*F + f].
// ---------------------------------------------------------------------------
__global__ __launch_bounds__(256) void k_attn(const int* __restrict__ adj,
                                              const float* __restrict__ g,
                                              const float* __restrict__ f1,
                                              const float* __restrict__ f2,
                                              const float* __restrict__ mcol,
                                              float* __restrict__ cat) {
  __shared__ float Ps[64 * LDT];
  __shared__ float Gs[64 * LDT];
  const int tid  = threadIdx.x;
  const int bh   = blockIdx.y, b = bh >> 2, hd = bh & 3;
  const int i0   = blockIdx.x * 64;
  const int wave = tid >> 5, lane = tid & 31;
  const int it   = wave & 3, ft0 = wave >> 2;
  const int mm   = lane & 15, kb = (lane >> 4) << 1;
  const int jlt  = tid & 63;         // column within chunk (fixed per thread)
  const int ilb  = tid >> 6;         // row sub-index (fixed per thread)
  const float* f1b = f1 + bh * N_;
  const float* f2b = f2 + bh * N_;
  const float* mb  = mcol + bh * N_;
  const float* gb  = g + (size_t)bh * N_ * F_;
  v8f acc0 = {}; v8f acc1 = {};

  float f1v[16];                     // block-row f1 values, fixed across chunks
#pragma unroll
  for (int r = 0; r < 16; ++r) f1v[r] = f1b[i0 + r * 4 + ilb];

  // G-copy mapping: 16B per lane per step
  const int gjl = tid >> 4;          // row 0..15 (+16 per step)
  const int gfl = (tid & 15) * 4;    // float4 column

  for (int jc = 0; jc < N_; jc += 64) {
    __syncthreads();

    // ---- stage G tile (pure copy; async DMA overlaps P generation) ----
#if USE_ASYNC_G
#pragma unroll
    for (int r = 0; r < 4; ++r) {
      const int jl = r * 16 + gjl;
      async_copy16(&gb[(size_t)(jc + jl) * F_ + gfl], &Gs[jl * LDT + gfl]);
    }
#else
#pragma unroll
    for (int r = 0; r < 4; ++r) {
      const int jl = r * 16 + gjl;
      const float4 v4 = *(const float4*)&gb[(size_t)(jc + jl) * F_ + gfl];
      *(float4*)&Gs[jl * LDT + gfl] = v4;
    }
#endif

    // ---- P tile: exp(mask(adj, leaky(f1+f2)) - m[j]) ----
    const float f2v = f2b[jc + jlt];
    const float mv  = mb[jc + jlt];
#pragma unroll
    for (int r = 0; r < 16; ++r) {
      const int il = r * 4 + ilb;
      const int a = adj[(size_t)(i0 + il) * N_ + jc + jlt];
      if ((r & 3) == 0 && jc + 64 < N_)
        __builtin_prefetch(&adj[(size_t)(i0 + il) * N_ + jc + 64 + jlt], 0, 3);
      float v = f1v[r] + f2v;
      v = fmaxf(v, ALPHA * v);
      const float e = a > 0 ? v : NEGV;
      Ps[il * LDT + jlt] = __expf(e - mv);
    }
#if USE_ASYNC_G
    WAIT_ASYNC();
#endif
    __syncthreads();

    // ---- 16x16x4 f32 WMMA over the chunk ----
    const float* pa = &Ps[(it * 16 + mm) * LDT];
#pragma unroll
    for (int kk = 0; kk < 16; ++kk) {
      const int k0 = kk * 4 + kb;
      const v2f a = *(const v2f*)&pa[k0];                     // ds_load_b64
      v2f b0 = { Gs[k0 * LDT + ft0 * 16 + mm], Gs[(k0 + 1) * LDT + ft0 * 16 + mm] };
      v2f b1 = { Gs[k0 * LDT + (ft0 + 2) * 16 + mm],
                 Gs[(k0 + 1) * LDT + (ft0 + 2) * 16 + mm] };
      acc0 = wmma_f32_16x16x4(a, b0, acc0);
      acc1 = wmma_f32_16x16x4(a, b1, acc1);
    }
  }
  const int mbse = (lane >> 4) << 3;
#pragma unroll
  for (int r = 0; r < 8; ++r) {
    const int mrow = it * 16 + mbse + r;
    float* crow = cat + (size_t)(b * N_ + i0 + mrow) * (H_ * F_) + hd * F_;
    crow[ft0 * 16 + mm]       = fmaxf(acc0[r], 0.f);   // ReLU (concat heads)
    crow[(ft0 + 2) * 16 + mm] = fmaxf(acc1[r], 0.f);
  }
}

// ---------------------------------------------------------------------------
// Kernel 5: out = leaky(cat @ Wl^T + bl).  cat: [B*N, 256], Wl: [64, 256].
// ---------------------------------------------------------------------------
__global__ __launch_bounds__(256) void k_final(const float* __restrict__ cat,
                                               const float* __restrict__ Wl,
                                               const float* __restrict__ bl,
                                               float* __restrict__ out) {
  __shared__ float As[64 * LDT];
  __shared__ float Bs[64 * LDT];
  const int tid  = threadIdx.x;
  const int row0 = blockIdx.x * 64;
  const int wave = tid >> 5, lane = tid & 31;
  const int it   = wave & 3, ft0 = wave >> 2;
  const int mm   = lane & 15, kb = (lane >> 4) << 1;
  v8f acc0 = {}; v8f acc1 = {};

  for (int kc = 0; kc < H_ * F_; kc += 64) {
    __syncthreads();
#pragma unroll
    for (int r = 0; r < 16; ++r) {
      const int idx = r * 256 + tid;
      const int rr = idx >> 6, cc = idx & 63;
      As[rr * LDT + cc] = cat[(size_t)(row0 + rr) * (H_ * F_) + kc + cc];
      Bs[cc * LDT + rr] = Wl[(size_t)rr * (H_ * F_) + kc + cc];  // Bs[c][f]=Wl[f][c]
    }
    __syncthreads();
    const float* pa = &As[(it * 16 + mm) * LDT];
#pragma unroll
    for (int kk = 0; kk < 16; ++kk) {
      const int k0 = kk * 4 + kb;
      const v2f a = *(const v2f*)&pa[k0];                     // ds_load_b64
      v2f b0 = { Bs[k0 * LDT + ft0 * 16 + mm], Bs[(k0 + 1) * LDT + ft0 * 16 + mm] };
      v2f b1 = { Bs[k0 * LDT + (ft0 + 2) * 16 + mm],
                 Bs[(k0 + 1) * LDT + (ft0 + 2) * 16 + mm] };
      acc0 = wmma_f32_16x16x4(a, b0, acc0);
      acc1 = wmma_f32_16x16x4(a, b1, acc1);
    }
  }
  const int mbse = (lane >> 4) << 3;
#pragma unroll
  for (int r = 0; r < 8; ++r) {
    const int mrow = it * 16 + mbse + r;
    const int n0 = ft0 * 16 + mm, n1 = (ft0 + 2) * 16 + mm;
    float v0 = acc0[r] + bl[n0];
    float v1 = acc1[r] + bl[n1];
    v0 = fmaxf(v0, ALPHA * v0);
    v1 = fmaxf(v1, ALPHA * v1);
    out[(size_t)(row0 + mrow) * F_ + n0] = v0;
    out[(size_t)(row0 + mrow) * F_ + n1] = v1;
  }
}

// ---------------------------------------------------------------------------
extern "C" void kernel_launch(void* const* d_in, const int* in_sizes, int n_in,
                              void* d_out, int out_size, void* d_ws,
                              size_t ws_size, hipStream_t stream) {
  (void)in_sizes; (void)n_in; (void)out_size; (void)ws_size;
  const float* x   = (const float*)d_in[0];
  const int*   adj = (const int*)d_in[1];
  const float* W   = (const float*)d_in[2];
  const float* a1  = (const float*)d_in[3];
  const float* a2  = (const float*)d_in[4];
  const float* Wl  = (const float*)d_in[5];
  const float* bl  = (const float*)d_in[6];
  float* out = (float*)d_out;

  // workspace layout (floats): h | f1 | f2 | m | rd | g | cat  (~24.5 MB)
  float* ws   = (float*)d_ws;
  float* h    = ws;                                  // BH*N*F
  float* f1   = h + (size_t)BH * N_ * F_;            // BH*N
  float* f2   = f1 + (size_t)BH * N_;                // BH*N
  float* mcol = f2 + (size_t)BH * N_;                // BH*N
  float* rd   = mcol + (size_t)BH * N_;              // BH*N
  float* g    = rd + (size_t)BH * N_;                // BH*N*F
  float* cat  = g + (size_t)BH * N_ * F_;            // B*N*(H*F)

  k_proj <<<dim3(N_ / 64, BH), 256, 0, stream>>>(x, W, h);
  k_fvec <<<(BH * N_) / 256, 256, 0, stream>>>(h, a1, a2, f1, f2);
  k_stats<<<dim3(N_ / 1024, BH), 256, 0, stream>>>(adj, f1, f2, mcol, rd);
  k_gmat <<<(BH * N_ * F_) / 1024, 256, 0, stream>>>(h, rd, g);
  k_attn <<<dim3(N_ / 64, BH), 256, 0, stream>>>(adj, g, f1, f2, mcol, cat);
  k_final<<<(B_ * N_) / 64, 256, 0, stream>>>(cat, Wl, bl, out);
}